// SpatioTemporalGNNLayer_9921374454080
// MI455X (gfx1250) — compile-verified
//
#include <hip/hip_runtime.h>

typedef float v2f __attribute__((ext_vector_type(2)));
typedef float v8f __attribute__((ext_vector_type(8)));

#define CCH    16          // channels
#define HW     256         // 16*16
#define CHW    4096        // 16*256
#define PAD_W  18
#define PAD_SZ 324         // 18*18
#define WTS    2304        // 16*16*9

#if defined(__HIP_DEVICE_COMPILE__) && __has_builtin(__builtin_amdgcn_global_load_async_to_lds_b128)
#define USE_ASYNC_LDS 1
#else
#define USE_ASYNC_LDS 0
#endif

#if USE_ASYNC_LDS
// Parameter types per hipcc diagnostic: (v4i __device__*, v4i __shared__*, Ii, Ii)
typedef int vi4 __attribute__((vector_size(16)));
typedef __attribute__((address_space(1))) vi4* gvi4p;   // global (cuda_device) v4i*
typedef __attribute__((address_space(3))) vi4* lvi4p;   // LDS (cuda_shared) v4i*

__device__ __forceinline__ gvi4p gbl_cast(const void* p) {
    return (gvi4p)(unsigned long long)p;
}
__device__ __forceinline__ lvi4p lds_cast(const void* p) {
    // flat LDS pointers carry the LDS offset in the low 32 bits (aperture is in [63:32])
    return (lvi4p)(unsigned int)(unsigned long long)p;
}
#endif

__global__ void zero_kernel(float4* __restrict__ out, float* __restrict__ deg,
                            int n_out4, int n_deg) {
    int i = blockIdx.x * blockDim.x + threadIdx.x;
    int stride = gridDim.x * blockDim.x;
    float4 z = make_float4(0.f, 0.f, 0.f, 0.f);
    for (int k = i; k < n_out4; k += stride) out[k] = z;
    for (int k = i; k < n_deg; k += stride) deg[k] = 0.0f;
}

__global__ void scatter_kernel(const float* __restrict__ x,
                               const long long* __restrict__ edge,
                               float* __restrict__ agg,
                               float* __restrict__ deg, int E) {
    int e = blockIdx.x;
    if (e >= E) return;
    int src = (int)edge[e];
    int dst = (int)edge[(size_t)E + e];
    if (threadIdx.x == 0) atomicAdd(&deg[dst], 1.0f);
    const float* xs = x + (size_t)src * CHW;
    float* ad = agg + (size_t)dst * CHW;
    for (int i = threadIdx.x; i < CHW; i += blockDim.x)
        atomicAdd(&ad[i], xs[i]);
}

__global__ __launch_bounds__(256)
void conv_wmma_kernel(const float* __restrict__ x,
                      const float* __restrict__ Ww, const float* __restrict__ Wb,
                      const float* __restrict__ Bw, const float* __restrict__ Bb,
                      const float* __restrict__ deg,
                      float* __restrict__ out /* holds agg (unnormalized) on entry */) {
    __shared__ float xs[CCH * PAD_SZ];      // x, zero-padded 18x18 halo per channel
    __shared__ float as[CCH * PAD_SZ];      // raw agg, zero-padded
    __shared__ float wws[WTS];              // W_w            [co][ci][tap]
    __shared__ float bws[WTS];              // B_w * invdeg   [co][ci][tap]
    __shared__ float bias_s[CCH];           // W_b + B_b

    const int n   = blockIdx.x;
    const int tid = threadIdx.x;
    const float* xN = x + (size_t)n * CHW;
    float* oN = out + (size_t)n * CHW;

    // degree normalization folded into the B-conv weights (conv is linear in W)
    const float invdeg = 1.0f / fmaxf(deg[n], 1.0f);

    // zero padded tiles, stage weights/bias
    for (int i = tid; i < CCH * PAD_SZ; i += 256) { xs[i] = 0.0f; as[i] = 0.0f; }
    for (int i = tid; i < WTS; i += 256) { wws[i] = Ww[i]; bws[i] = Bw[i] * invdeg; }
    if (tid < CCH) bias_s[tid] = Wb[tid] + Bb[tid];
    __syncthreads();   // halo zeros visible before interior fill

#if USE_ASYNC_LDS
    // Async DMA global -> LDS (16B per lane), scattered into the padded layout.
    // 1024 x b128 chunks per array: chunk c covers floats [4c, 4c+4) of the image,
    // i.e. channel c>>6, row (c>>2)&15, quarter c&3.
    for (int c = tid; c < 1024; c += 256) {
        int ci = c >> 6;
        int rh = (c >> 2) & 15;
        int q  = c & 3;
        int lo = ci * PAD_SZ + (rh + 1) * PAD_W + 1 + q * 4;
        __builtin_amdgcn_global_load_async_to_lds_b128(
            gbl_cast(xN + c * 4), lds_cast(xs + lo), 0, 0);
        __builtin_amdgcn_global_load_async_to_lds_b128(
            gbl_cast(oN + c * 4), lds_cast(as + lo), 0, 0);
    }
#if __has_builtin(__builtin_amdgcn_s_wait_asynccnt)
    __builtin_amdgcn_s_wait_asynccnt(0);
#else
    asm volatile("s_wait_asynccnt 0" ::: "memory");
#endif
#else
    for (int i = tid; i < CHW; i += 256) {
        int ci = i >> 8;
        int p  = i & 255;
        int h  = p >> 4, w = p & 15;
        int pp = ci * PAD_SZ + (h + 1) * PAD_W + (w + 1);
        xs[pp] = xN[i];
        as[pp] = oN[i];
    }
#endif
    __syncthreads();

    const int lane  = tid & 31;
    const int wave  = tid >> 5;
    const int half  = lane >> 4;     // 0: lanes 0-15, 1: lanes 16-31
    const int wcol  = lane & 15;     // cout row (A) / w column (B,D)
    const int hBase = wave * 2;      // this wave owns output rows hBase, hBase+1

    v8f acc0 = {};
    v8f acc1 = {};

    for (int conv = 0; conv < 2; ++conv) {
        const float* in_s = conv ? as  : xs;
        const float* w_s  = conv ? bws : wws;
        #pragma unroll
        for (int tap = 0; tap < 9; ++tap) {
            const int dy = tap / 3, dx = tap % 3;
            #pragma unroll
            for (int kb = 0; kb < 4; ++kb) {
                // A 16x4 f32 layout: v0 holds K = kb*4 + {0|2}, v1 holds K = kb*4 + {1|3}
                const int cinA = kb * 4 + half * 2;
                const int cinB = cinA + 1;
                v2f a;
                a.x = w_s[wcol * (CCH * 9) + cinA * 9 + tap];
                a.y = w_s[wcol * (CCH * 9) + cinB * 9 + tap];
                {   // row hBase
                    int pos = (hBase + dy) * PAD_W + (wcol + dx);
                    v2f b;
                    b.x = in_s[cinA * PAD_SZ + pos];
                    b.y = in_s[cinB * PAD_SZ + pos];
                    acc0 = __builtin_amdgcn_wmma_f32_16x16x4_f32(
                        false, a, false, b, (short)0, acc0, false, false);
                }
                {   // row hBase+1
                    int pos = (hBase + 1 + dy) * PAD_W + (wcol + dx);
                    v2f b;
                    b.x = in_s[cinA * PAD_SZ + pos];
                    b.y = in_s[cinB * PAD_SZ + pos];
                    acc1 = __builtin_amdgcn_wmma_f32_16x16x4_f32(
                        false, a, false, b, (short)0, acc1, false, false);
                }
            }
        }
    }

    // epilogue: bias + relu + store. D layout: VGPR r -> co = r + 8*half, n = wcol
    #pragma unroll
    for (int r = 0; r < 8; ++r) {
        int co = r + half * 8;
        float bv = bias_s[co];
        float v0 = fmaxf(acc0[r] + bv, 0.0f);
        float v1 = fmaxf(acc1[r] + bv, 0.0f);
        oN[co * HW + hBase * 16 + wcol]       = v0;
        oN[co * HW + (hBase + 1) * 16 + wcol] = v1;
    }
}

extern "C" void kernel_launch(void* const* d_in, const int* in_sizes, int n_in,
                              void* d_out, int out_size, void* d_ws, size_t ws_size,
                              hipStream_t stream) {
    const float*     x    = (const float*)d_in[0];
    const long long* edge = (const long long*)d_in[1];
    const float*     Ww   = (const float*)d_in[2];
    const float*     Wb   = (const float*)d_in[3];
    const float*     Bw   = (const float*)d_in[4];
    const float*     Bb   = (const float*)d_in[5];
    float* out = (float*)d_out;
    float* deg = (float*)d_ws;          // 4096 floats of scratch

    const int E      = in_sizes[1] / 2;
    const int nNodes = in_sizes[0] / CHW;

    zero_kernel<<<1024, 256, 0, stream>>>((float4*)out, deg, nNodes * CHW / 4, nNodes);
    scatter_kernel<<<E, 256, 0, stream>>>(x, edge, out, deg, E);
    conv_wmma_kernel<<<nNodes, 256, 0, stream>>>(x, Ww, Wb, Bw, Bb, deg, out);
}